// TemporalGNN_26001732010513
// MI455X (gfx1250) — compile-verified
//
#include <hip/hip_runtime.h>

typedef float v2f __attribute__((ext_vector_type(2)));
typedef float v4f __attribute__((ext_vector_type(4)));
typedef float v8f __attribute__((ext_vector_type(8)));

#define NN 50000
#define TT 64
#define FF 64
#define EE 800000
#define WPAD 68   // LDS column stride for transposed W: 68 ≡ 4 banks/col -> conflict-free b64 reads

// ---------------- small setup kernels ----------------

__global__ void k_init_deg(float* __restrict__ deg) {
    int i = blockIdx.x * blockDim.x + threadIdx.x;
    if (i < NN) deg[i] = 1.0f;   // self-loop
}

__global__ void k_zero_out(float4* __restrict__ out) {
    // exact: N*T*F/4 = 51,200,000 = 200,000 blocks * 256 threads
    long long i = (long long)blockIdx.x * blockDim.x + threadIdx.x;
    out[i] = make_float4(0.f, 0.f, 0.f, 0.f);
}

__global__ void k_deg_accum(const int* __restrict__ dst, float* __restrict__ deg) {
    int e = blockIdx.x * blockDim.x + threadIdx.x;
    if (e < EE) atomicAdd(&deg[dst[e]], 1.0f);
}

__global__ void k_dis(const float* __restrict__ deg, float* __restrict__ dis) {
    int i = blockIdx.x * blockDim.x + threadIdx.x;
    if (i < NN) dis[i] = rsqrtf(deg[i]);
}

__global__ void k_norm(const int* __restrict__ src, const int* __restrict__ dst,
                       const float* __restrict__ dis, float* __restrict__ norm) {
    int e = blockIdx.x * blockDim.x + threadIdx.x;
    if (e < EE) norm[e] = dis[src[e]] * dis[dst[e]];
}

// ---------------- WMMA GEMM: xw[r,:] = x[r,:] @ W, r = n*T + t ----------------
// (N*T x 64) * (64 x 64), fp32 via V_WMMA_F32_16X16X4_F32 (matches the fp32 reference;
// the problem is memory/atomic bound so there is no payoff in dropping precision).
// A 16x4 f32 layout: lane -> M = lane&15; VGPR v holds K = 2*(lane>>4) + v.
// B 4x16 f32 layout: lane -> N = lane&15; VGPR v holds K = 2*(lane>>4) + v.
// C 16x16 f32 layout: VGPR v: lanes 0-15 -> M=v, lanes 16-31 -> M=v+8; N = lane&15.

__global__ __launch_bounds__(256) void k_gemm_wmma(const float* __restrict__ x,
                                                   const float* __restrict__ W,
                                                   float* __restrict__ xw) {
    // W transposed in LDS: Wt[col*WPAD + k] = W[k*FF + col]
    // -> each lane's B fragment is one aligned, bank-conflict-free ds_load_b64.
    __shared__ float Wt[FF * WPAD];        // ~17 KB of the 320 KB/WGP LDS
    for (int i = threadIdx.x; i < FF * FF; i += 256) {
        const int k   = i >> 6;            // i / 64
        const int col = i & 63;            // i % 64
        Wt[col * WPAD + k] = W[i];
    }
    __syncthreads();

    const int wave = threadIdx.x >> 5;     // wave32: 8 waves / block
    const int lane = threadIdx.x & 31;
    const int tile = blockIdx.x * 8 + wave;            // 16-row tile id
    const long long rowBase = (long long)tile * 16;

    const int m    = lane & 15;
    const int half = lane >> 4;

    // Preload all 16 A fragments (K=0..63 in steps of 4) for this wave's 16 rows.
    v2f a[16];
    const float* xr = x + (rowBase + m) * FF + 2 * half;
#pragma unroll
    for (int ks = 0; ks < 16; ++ks) {
        float2 v = *(const float2*)(xr + 4 * ks);      // K = 4*ks + 2*half + {0,1}
        a[ks][0] = v.x;
        a[ks][1] = v.y;
    }

    float* orow = xw + rowBase * FF;
#pragma unroll
    for (int nt = 0; nt < 4; ++nt) {                   // 4 x 16 output columns
        v8f c = {};
        const float* bcol = Wt + (nt * 16 + m) * WPAD + 2 * half;
#pragma unroll
        for (int ks = 0; ks < 16; ++ks) {
            v2f b = *(const v2f*)(bcol + 4 * ks);      // contiguous K pair: single ds_load_b64
            c = __builtin_amdgcn_wmma_f32_16x16x4_f32(
                    /*neg_a=*/false, a[ks], /*neg_b=*/false, b,
                    /*c_mod=*/(short)0, c, /*reuse_a=*/false, /*reuse_b=*/false);
        }
#pragma unroll
        for (int v = 0; v < 8; ++v) {
            const int row = half * 8 + v;
            orow[row * FF + nt * 16 + m] = c[v];
        }
    }
}

// ---------------- edge scatter: out[dst,t,:] += norm[e] * xw[src,t,:] ----------------
// t-major wave ordering so each timestep's xw/agg slices (~26 MB) live in the 192 MB L2.

__global__ __launch_bounds__(256) void k_scatter(const float* __restrict__ xw,
                                                 const int* __restrict__ src,
                                                 const int* __restrict__ dst,
                                                 const float* __restrict__ norm,
                                                 float* __restrict__ out) {
    const long long wid = (long long)blockIdx.x * 8 + (threadIdx.x >> 5); // one wave = one (e,t)
    const int lane = threadIdx.x & 31;
    const int e = (int)(wid % EE);
    const int t = (int)(wid / EE);

    const int   s = src[e];
    const int   d = dst[e];
    const float w = norm[e];

    const float2 v = *(const float2*)(xw + ((long long)s * TT + t) * FF + lane * 2);
    float* o = out + ((long long)d * TT + t) * FF + lane * 2;
    atomicAdd(o,     w * v.x);
    atomicAdd(o + 1, w * v.y);
}

// ---------------- finalize: out = relu(out + dis^2 * xw + b) ----------------
// xw and the accumulated out are each read exactly once here -> non-temporal loads;
// the final out store is streaming -> non-temporal store. Keeps L2 clean.

__global__ __launch_bounds__(256) void k_finalize(const float* __restrict__ xw,
                                                  const float* __restrict__ dis,
                                                  const float* __restrict__ bias,
                                                  float* __restrict__ out) {
    const long long g4 = (long long)blockIdx.x * 256 + threadIdx.x;  // float4 index, exact grid
    const long long g  = g4 * 4;
    const int n = (int)(g / (TT * FF));
    const int f = (int)(g & (FF - 1));

    const float d  = dis[n];
    const float sn = d * d;                      // 1/deg
    const v4f xv = __builtin_nontemporal_load((const v4f*)(xw + g));
    const v4f ov = __builtin_nontemporal_load((const v4f*)(out + g));
    const v4f bv = *(const v4f*)(bias + f);

    v4f r;
    r.x = fmaxf(ov.x + sn * xv.x + bv.x, 0.0f);
    r.y = fmaxf(ov.y + sn * xv.y + bv.y, 0.0f);
    r.z = fmaxf(ov.z + sn * xv.z + bv.z, 0.0f);
    r.w = fmaxf(ov.w + sn * xv.w + bv.w, 0.0f);
    __builtin_nontemporal_store(r, (v4f*)(out + g));
}

// ---------------- launcher ----------------

extern "C" void kernel_launch(void* const* d_in, const int* in_sizes, int n_in,
                              void* d_out, int out_size, void* d_ws, size_t ws_size,
                              hipStream_t stream) {
    const float* x  = (const float*)d_in[0];   // [N,T,F] f32
    const int*   ei = (const int*)d_in[1];     // [2,E]   i32
    const float* W  = (const float*)d_in[2];   // [F,F]   f32
    const float* b  = (const float*)d_in[3];   // [F]     f32
    float*       out = (float*)d_out;          // [N,T,F] f32 (agg accumulates here)

    const int* src = ei;
    const int* dst = ei + EE;

    // workspace layout
    float* xw   = (float*)d_ws;                          // N*T*F  (819.2 MB)
    float* deg  = xw  + (size_t)NN * TT * FF;            // N
    float* dis  = deg + NN;                              // N
    float* norm = dis + NN;                              // E

    k_init_deg <<<(NN + 255) / 256, 256, 0, stream>>>(deg);
    k_zero_out <<<(NN * TT * FF / 4) / 256, 256, 0, stream>>>((float4*)out);
    k_deg_accum<<<(EE + 255) / 256, 256, 0, stream>>>(dst, deg);
    k_dis      <<<(NN + 255) / 256, 256, 0, stream>>>(deg, dis);
    k_norm     <<<(EE + 255) / 256, 256, 0, stream>>>(src, dst, dis, norm);

    k_gemm_wmma<<<(NN * TT / 16) / 8, 256, 0, stream>>>(x, W, xw);       // 25,000 blocks

    k_scatter  <<<(unsigned)((long long)EE * TT / 8), 256, 0, stream>>>(xw, src, dst, norm, out);

    k_finalize <<<(NN * TT * FF / 4) / 256, 256, 0, stream>>>(xw, dis, b, out);
}